// RNN_Model_53412213293313
// MI455X (gfx1250) — compile-verified
//
#include <hip/hip_runtime.h>

typedef __attribute__((ext_vector_type(2))) float v2f;
typedef __attribute__((ext_vector_type(8))) float v8f;

#define EP 112            // E=100 padded to 7 WMMA tiles of 16
#define KSTEPS 28         // EP / 4 (K per fp32 WMMA)
#define NT 7              // N tiles (112 / 16)
#define NW 8              // waves per block in the level kernel

// ---------------------------------------------------------------------------
// Kernel 1: gather leaf embeddings into padded h0 buffer (pad cols = 0.0f so
// WMMA never sees non-finite data in the K padding).
// ---------------------------------------------------------------------------
__global__ void gather_kernel(const int* __restrict__ ids,
                              const float* __restrict__ emb,
                              float* __restrict__ h, int L) {
  size_t t = (size_t)blockIdx.x * blockDim.x + threadIdx.x;
  size_t total = (size_t)L * EP;
  if (t >= total) return;
  int i = (int)(t / EP);
  int k = (int)(t % EP);
  h[t] = (k < 100) ? emb[(size_t)ids[i] * 100 + k] : 0.0f;
}

// ---------------------------------------------------------------------------
// Kernel 2: per-level projection  out[pos(lvl,j)] = h[j] @ Wp^T + bp
// One wave32 per node; coalesced row reads; shfl_xor tree reduction.
// pos closed form: (j << (lvl+1)) - popcount(j) + 2^(lvl+1) - 2
// ---------------------------------------------------------------------------
__global__ void proj_kernel(const float* __restrict__ h,
                            const float* __restrict__ Wp,
                            const float* __restrict__ bp,
                            float* __restrict__ out, int count, int lvl) {
  int wave = (int)((blockIdx.x * (size_t)blockDim.x + threadIdx.x) >> 5);
  int lane = threadIdx.x & 31;
  if (wave >= count) return;
  const float* row = h + (size_t)wave * EP;
  float a0 = 0.0f, a1 = 0.0f;
  for (int k = lane; k < 100; k += 32) {
    float v = row[k];
    a0 += v * Wp[k];
    a1 += v * Wp[100 + k];
  }
  for (int off = 16; off > 0; off >>= 1) {
    a0 += __shfl_xor(a0, off, 32);
    a1 += __shfl_xor(a1, off, 32);
  }
  if (lane == 0) {
    int pos = (wave << (lvl + 1)) - __popc(wave) + (1 << (lvl + 1)) - 2;
    out[(size_t)pos * 2 + 0] = a0 + bp[0];
    out[(size_t)pos * 2 + 1] = a1 + bp[1];
  }
}

// ---------------------------------------------------------------------------
// Kernel 3: one tree level.  h_next[i] = relu((h[2i]+h[2i+1]) @ Wl^T + 2*bl)
// fp32 WMMA 16x16x4. Per wave: one 16-node M-tile, all 7 N-tiles at once.
//   A (16x4 f32): lane<16 -> M=lane,K={0,1}; lane>=16 -> M=lane-16,K={2,3}
//   B (4x16 f32): lane<16 -> K={0,1},N=lane; lane>=16 -> K={2,3},N=lane-16
//   D (16x16)   : VGPR r -> M = r + 8*(lane>=16), N = lane&15
// K-outer loop: each A pair is loaded once (b64) and feeds 7 WMMAs into 7
// independent accumulators (56 VGPRs) -> A reuse without a long-lived array,
// and 7 interleaved D->C chains for WMMA pipelining.  B is staged K-pair-
// interleaved in LDS (WlP[k/2][n][2] = {Wl[n][k], Wl[n][k+1]}) so each B
// operand is one aligned ds_load_b64 with no repacking movs.
// ---------------------------------------------------------------------------
__global__ void __launch_bounds__(NW * 32)
level_kernel(const float* __restrict__ hin, float* __restrict__ hout,
             const float* __restrict__ Wl, const float* __restrict__ bl,
             int count) {
  extern __shared__ float smem[];
  float* WlP   = smem;             // [KSTEPS*2][EP][2] pair-interleaved Wl^T
  float* blS   = WlP + EP * EP;    // [EP]
  float* sBase = blS + EP;         // [NW][16][EP] pair sums

  const int tid  = threadIdx.x;
  const int lane = tid & 31;
  const int wave = tid >> 5;

  // Stage Wl^T pair-interleaved (+ zero pad) and bl (block-cooperative).
  // WlP[((k>>1)*EP + e)*2 + (k&1)] = Wl[e][k]
  for (int idx = tid; idx < EP * EP; idx += blockDim.x) {
    int k = idx / EP, e = idx % EP;
    float v = (k < 100 && e < 100) ? Wl[e * 100 + k] : 0.0f;
    WlP[(((k >> 1) * EP) + e) * 2 + (k & 1)] = v;
  }
  for (int idx = tid; idx < EP; idx += blockDim.x)
    blS[idx] = (idx < 100) ? bl[idx] : 0.0f;

  // Stage this wave's pair-summed A tile: s[m][k] = h[2i][k] + h[2i+1][k].
  const int mbase = (blockIdx.x * NW + wave) * 16;
  float* s = sBase + wave * 16 * EP;
  for (int idx = lane; idx < 16 * EP; idx += 32) {
    int m = idx / EP, k = idx % EP;
    int node = mbase + m;
    float v = 0.0f;
    if (node < count) {
      size_t r = (size_t)(2 * node) * EP + k;
      v = hin[r] + hin[r + EP];
    }
    s[idx] = v;
  }
  __syncthreads();
  if (mbase >= count) return;  // wave-uniform exit; EXEC all-ones for WMMA

  const int hi = lane >> 4;   // 0/1: selects K+2 (A,B) and M+8 (D)
  const int lo = lane & 15;

  v8f acc[NT];
#pragma unroll
  for (int nt = 0; nt < NT; ++nt) acc[nt] = (v8f){};

#pragma unroll
  for (int kk = 0; kk < KSTEPS; ++kk) {
    const int ka = 4 * kk + 2 * hi;
    v2f a = *(const v2f*)&s[lo * EP + ka];           // A[M=lo][ka, ka+1]
    const int pr = 2 * kk + hi;                      // B pair row index
#pragma unroll
    for (int nt = 0; nt < NT; ++nt) {
      v2f b = *(const v2f*)&WlP[(pr * EP + nt * 16 + lo) * 2];
      acc[nt] = __builtin_amdgcn_wmma_f32_16x16x4_f32(
          false, a, false, b, (short)0, acc[nt], false, false);
    }
  }

  // Epilogue: D layout VGPR r -> M = r + 8*hi, N = lane&15.
#pragma unroll
  for (int nt = 0; nt < NT; ++nt) {
    const int n = nt * 16 + lo;
    const float tb = 2.0f * blS[n];
#pragma unroll
    for (int r = 0; r < 8; ++r) {
      int node = mbase + r + 8 * hi;
      if (node < count) {
        float v = acc[nt][r] + tb;
        hout[(size_t)node * EP + n] = v > 0.0f ? v : 0.0f;
      }
    }
  }
}

// ---------------------------------------------------------------------------
// Host driver: gather, then per level {proj, level-step}, ping-pong h buffers
// in d_ws (needs (L + L/2) * 112 * 4 bytes ~= 88 MB).
// ---------------------------------------------------------------------------
extern "C" void kernel_launch(void* const* d_in, const int* in_sizes, int n_in,
                              void* d_out, int out_size, void* d_ws, size_t ws_size,
                              hipStream_t stream) {
  const int*   word_ids = (const int*)  d_in[0];
  const float* emb      = (const float*)d_in[1];
  const float* Wl       = (const float*)d_in[2];
  const float* bl       = (const float*)d_in[3];
  const float* Wp       = (const float*)d_in[4];
  const float* bp       = (const float*)d_in[5];
  float* out = (float*)d_out;

  const int L = in_sizes[0];
  int D = 0;
  while ((1 << D) < L) ++D;

  float* bufA = (float*)d_ws;                 // even levels (max L rows)
  float* bufB = bufA + (size_t)L * EP;        // odd levels  (max L/2 rows)

  {
    size_t total = (size_t)L * EP;
    unsigned blocks = (unsigned)((total + 255) / 256);
    gather_kernel<<<blocks, 256, 0, stream>>>(word_ids, emb, bufA, L);
  }

  const size_t smem = (size_t)(EP * EP + EP + NW * 16 * EP) * sizeof(float);

  for (int lvl = 0; lvl <= D; ++lvl) {
    const int count = L >> lvl;
    float* h = (lvl & 1) ? bufB : bufA;

    {
      int wavesPerBlock = 8;
      int blocks = (count + wavesPerBlock - 1) / wavesPerBlock;
      proj_kernel<<<blocks, 32 * wavesPerBlock, 0, stream>>>(h, Wp, bp, out,
                                                             count, lvl);
    }

    if (lvl < D) {
      float* hn = (lvl & 1) ? bufA : bufB;
      const int cn = count >> 1;
      const int nodesPerBlock = NW * 16;
      int blocks = (cn + nodesPerBlock - 1) / nodesPerBlock;
      level_kernel<<<blocks, 32 * NW, smem, stream>>>(h, hn, Wl, bl, cn);
    }
  }
}